// Q2n_1898375545542
// MI455X (gfx1250) — compile-verified
//
#include <hip/hip_runtime.h>
#include <math.h>

// ---------------- CDNA5 vector types ----------------
typedef __attribute__((ext_vector_type(2))) float v2f;
typedef __attribute__((ext_vector_type(8))) float v8f;

#define C_CH 16
#define HW   512
#define QS   32

// ---------------- compile-time sedenion sign table ----------------
// Runs the reference _cd_product recursion symbolically on basis vectors.
// e_i * e_j = sign * e_(i^j); we store sign in s[i][j].
struct CDTable { float s[16][16]; };

constexpr void cdp_c(const int* o1, const int* o2, int* out, int n) {
    if (n == 1) { out[0] = o1[0] * o2[0]; return; }
    const int m = n / 2;
    int a[8] = {}, b[8] = {}, c[8] = {}, d[8] = {}, bs[8] = {}, as[8] = {};
    for (int t = 0; t < m; ++t) {
        const int s = (t == 0) ? 1 : -1;
        a[t] = o1[t];         b[t] = o1[m + t] * s;
        c[t] = o2[t];         d[t] = o2[m + t] * s;
        bs[t] = b[t] * s;     as[t] = a[t] * s;
    }
    if (n == 2) {
        out[0] = a[0] * c[0] - d[0] * b[0];
        out[1] = a[0] * d[0] + c[0] * b[0];
        return;
    }
    int r1[8] = {}, r2[8] = {}, r3[8] = {}, r4[8] = {};
    cdp_c(a, c, r1, m);
    cdp_c(d, bs, r2, m);
    cdp_c(as, d, r3, m);
    cdp_c(c, b, r4, m);
    for (int t = 0; t < m; ++t) { out[t] = r1[t] - r2[t]; out[m + t] = r3[t] + r4[t]; }
}

constexpr CDTable make_table() {
    CDTable T{};
    for (int i = 0; i < 16; ++i)
        for (int j = 0; j < 16; ++j) {
            int o1[16] = {}, o2[16] = {}, out[16] = {};
            o1[i] = 1; o2[j] = 1;
            cdp_c(o1, o2, out, 16);
            T.s[i][j] = (float)out[i ^ j];
        }
    return T;
}

__constant__ CDTable g_cd = make_table();

// ---------------- wave helpers (wave32) ----------------
__device__ inline float wave_sum32(float v) {
    #pragma unroll
    for (int off = 16; off > 0; off >>= 1) v += __shfl_xor(v, off, 32);
    return v;
}

// ---------------- per-block kernel: one wave32 per 8x8x16 block ----------------
// Lane L: channel c = L&15, pixel half t = L>>4 (owns pixels with p%4 in {2t,2t+1}).
// G[i][j] = sum_p im1n[i,p]*im2n[j,p] via 16x V_WMMA_F32_16X16X4_F32.
__global__ __launch_bounds__(32) void q2n_block_kernel(
    const float* __restrict__ labels,   // im1 (statistics source)
    const float* __restrict__ outputs,  // im2
    float* __restrict__ q2n_out)        // [2048] per-block q2n
{
    const int lane = threadIdx.x;
    const int c = lane & 15;
    const int t = lane >> 4;
    const int blk = blockIdx.x;              // ((b*16 + sx)*16 + sy)
    const int b  = blk >> 8;
    const int sx = (blk >> 4) & 15;
    const int sy = blk & 15;

    const size_t chanBase =
        ((size_t)(b * C_CH + c) * HW + (size_t)sx * QS) * HW + (size_t)sy * QS;

    // ---- load + round: 32 pixels of this channel per lane (pairs are contiguous) ----
    float x1[32], x2[32];
    #pragma unroll
    for (int g = 0; g < 16; ++g) {
        const int p = 4 * g + 2 * t;         // pixel index 0..63
        const int i = p >> 3;                // row in 8x8
        const int j = p & 7;                 // col in 8x8 (even)
        const size_t off = chanBase + (size_t)i * HW + (size_t)j;
        const float2 L2 = *(const float2*)(labels + off);
        const float2 O2 = *(const float2*)(outputs + off);
        x1[2 * g]     = rintf(L2.x);  x1[2 * g + 1] = rintf(L2.y);
        x2[2 * g]     = rintf(O2.x);  x2[2 * g + 1] = rintf(O2.y);
    }

    // ---- per-channel mean / std (ddof=1) of im1 ----
    float s1 = 0.0f, s2 = 0.0f;
    #pragma unroll
    for (int r = 0; r < 32; ++r) { s1 += x1[r]; s2 += x1[r] * x1[r]; }
    s1 += __shfl_xor(s1, 16, 32);
    s2 += __shfl_xor(s2, 16, 32);
    const float m   = s1 * (1.0f / 64.0f);
    float var = (s2 - s1 * s1 * (1.0f / 64.0f)) * (1.0f / 63.0f);
    var = fmaxf(var, 0.0f);
    float sd = sqrtf(var);
    sd = (sd == 0.0f) ? 1e-10f : sd;
    const float inv_sd = 1.0f / sd;

    const float m0 = __shfl(m, 0, 32);       // channel-0 mean
    const bool  cond = (m0 == 0.0f);
    const float negc = (c == 0) ? 1.0f : -1.0f;

    // ---- normalize; accumulate channel sums and global sum-of-squares ----
    float sm1 = 0.0f, sm2 = 0.0f, sq1 = 0.0f, sq2 = 0.0f;
    #pragma unroll
    for (int r = 0; r < 32; ++r) {
        const float a  = (x1[r] - m) * inv_sd + 1.0f;
        float bb = cond ? (x2[r] - m + 1.0f) : ((x2[r] - m) * inv_sd + 1.0f);
        bb *= negc;
        x1[r] = a;  x2[r] = bb;
        sm1 += a;   sm2 += bb;
        sq1 += a * a; sq2 += bb * bb;
    }
    sm1 += __shfl_xor(sm1, 16, 32);
    sm2 += __shfl_xor(sm2, 16, 32);
    const float m1c = sm1 * (1.0f / 64.0f);
    const float m2c = sm2 * (1.0f / 64.0f);

    const float tq1 = wave_sum32(sq1);                 // Σ_{c,p} im1n²
    const float tq2 = wave_sum32(sq2);                 // Σ_{c,p} im2n²
    const float mq1m_sq = wave_sum32(m1c * m1c) * 0.5f; // Σ_c m1²  (each channel twice)
    const float mq2m_sq = wave_sum32(m2c * m2c) * 0.5f;

    const float temp  = 64.0f / 63.0f;
    const float int1  = temp * tq1 * (1.0f / 64.0f);
    const float int2  = temp * tq2 * (1.0f / 64.0f);
    const float term4 = mq1m_sq + mq2m_sq;
    const float term2 = sqrtf(mq1m_sq) * sqrtf(mq2m_sq);
    const float term3 = int1 + int2 - temp * term4;
    const float mean_bias = 2.0f * term2 / term4;
    const bool  cond3 = (term3 == 0.0f);
    const float cbm = 2.0f / (cond3 ? 1.0f : term3);

    // ---- G = im1n · im2nᵀ  (16x16, K=64) via f32 WMMA, K=4 per step ----
    // A layout: lane -> M=c, K-pair {2t,2t+1} of each K-base 4*kk (== x1[2kk..]).
    // B layout (mirrored): lane -> N=c, same K-pair (== x2[2kk..]).
    v8f acc = {};
    #pragma unroll
    for (int kk = 0; kk < 16; ++kk) {
        v2f av = { x1[2 * kk], x1[2 * kk + 1] };
        v2f bv = { x2[2 * kk], x2[2 * kk + 1] };
        acc = __builtin_amdgcn_wmma_f32_16x16x4_f32(
            /*neg_a=*/false, av, /*neg_b=*/false, bv,
            /*c_mod=*/(short)0, acc, /*reuse_a=*/false, /*reuse_b=*/false);
    }

    // ---- spill G + channel means to LDS, contract with sedenion sign table ----
    __shared__ float Gs[16][16];
    __shared__ float m1s[16], m2s[16];
    #pragma unroll
    for (int v = 0; v < 8; ++v) Gs[v + 8 * t][c] = acc[v];  // D: M=v+8t, N=c
    if (t == 0) { m1s[c] = m1c; m2s[c] = m2c; }
    __syncthreads();

    const int k = c;                  // lanes t=0 and t=1 compute same k; t=1 zeroed below
    float qv = 0.0f, qm = 0.0f;
    #pragma unroll
    for (int i = 0; i < 16; ++i) {
        const int j = i ^ k;
        const float s = g_cd.s[i][j];
        qv += s * Gs[i][j];
        qm += s * m1s[i] * m2s[j];
    }
    qv *= temp * (1.0f / 64.0f);
    const float qmain = (qv - temp * qm) * (mean_bias * cbm);
    const float qk = cond3 ? ((k == 15) ? mean_bias : 0.0f) : qmain;
    // NOTE: reference's where(q==0, 1e-30) is irrelevant in f32 (1e-60 underflows).
    const float qq  = (t == 0) ? qk * qk : 0.0f;
    const float tot = wave_sum32(qq);
    if (lane == 0) q2n_out[blk] = sqrtf(tot);
}

// ---------------- deterministic final reduction: out[b] = mean over 256 blocks ----------------
__global__ void q2n_reduce_kernel(const float* __restrict__ q2n, float* __restrict__ out) {
    const int b = threadIdx.x;
    if (b >= 8) return;
    float s = 0.0f;
    for (int i = 0; i < 256; ++i) s += q2n[b * 256 + i];  // fixed order -> deterministic
    out[b] = s * (1.0f / 256.0f);
}

extern "C" void kernel_launch(void* const* d_in, const int* in_sizes, int n_in,
                              void* d_out, int out_size, void* d_ws, size_t ws_size,
                              hipStream_t stream) {
    (void)in_sizes; (void)n_in; (void)out_size; (void)ws_size;
    const float* outputs = (const float*)d_in[0];  // im2
    const float* labels  = (const float*)d_in[1];  // im1 (statistics source)
    float* q2n_scratch = (float*)d_ws;             // 2048 floats
    float* out = (float*)d_out;                    // 8 floats

    q2n_block_kernel<<<2048, 32, 0, stream>>>(labels, outputs, q2n_scratch);
    q2n_reduce_kernel<<<1, 8, 0, stream>>>(q2n_scratch, out);
}